// AttnNeck_6030134083913
// MI455X (gfx1250) — compile-verified
//
#include <hip/hip_runtime.h>
#include <hip/hip_bf16.h>
#include <cstdint>

typedef __attribute__((ext_vector_type(2))) float v2f;
typedef __attribute__((ext_vector_type(8))) float v8f;

#define CCH 256
#define HH  64
#define WW  64
#define HWP 4096

__device__ __forceinline__ v8f wmma_f32x4(v2f a, v2f b, v8f c) {
    // D(16x16,f32) = A(16x4,f32) x B(4x16,f32) + C
    return __builtin_amdgcn_wmma_f32_16x16x4_f32(
        /*neg_a=*/false, a, /*neg_b=*/false, b,
        /*c_mod=*/(short)0, c, /*reuse_a=*/false, /*reuse_b=*/false);
}

__device__ __forceinline__ v2f ld2(const float* p) { return *(const v2f*)p; }
__device__ __forceinline__ void st2(float* p, v2f v) { *(v2f*)p = v; }

// ---------------------------------------------------------------------------
// Conv 3x3 (stride 1, SAME, no bias) + ReLU, NCHW, C_in = C_out = 256.
// Implicit GEMM; K-dim (ci) innermost in LDS so all WMMA fragments are
// contiguous 8B-aligned pairs -> ds_load_b64 straight into operand regs.
// Block: 256 thr (8 waves) -> 64 co x 64 pixels (one image row).
// ---------------------------------------------------------------------------
#define CI_STR 18   // 16 ci padded to 18 (even => b64 aligned; stride 18 words
                    // keeps 16 lanes on distinct banks)

__global__ __launch_bounds__(256) void conv3x3_relu_kernel(
    const float* __restrict__ in, const float* __restrict__ w,
    float* __restrict__ out)
{
    __shared__ float in_t[3 * 72 * CI_STR];  // [row3][x72][ci18]  15.6KB
    __shared__ float w_t[9 * 64 * CI_STR];   // [tap][co64][ci18]  41.5KB

    const int tid  = threadIdx.x;
    const int lane = tid & 31;
    const int wid  = tid >> 5;
    const int half = lane >> 4;
    const int l16  = lane & 15;

    const int co0 = blockIdx.x * 64;
    const int h   = blockIdx.y;
    const int b   = blockIdx.z;
    const float* inb = in + (size_t)b * CCH * HWP;

    const int c_s = wid >> 1;        // co subtile 0..3
    const int p_s = (wid & 1) * 2;   // pixel subtile base {0,2}

    v8f acc0 = {}; v8f acc1 = {};

    for (int ci0 = 0; ci0 < CCH; ci0 += 16) {
        __syncthreads();
        // Stage input rows h-1..h+1, 16 ci, zero-padded width; ci innermost.
        for (int idx = tid; idx < 16 * 3 * 72; idx += 256) {
            int ci  = idx / 216;
            int rem = idx - ci * 216;
            int r   = rem / 72;
            int x   = rem - r * 72;
            int hp  = h + r - 1;
            float v = 0.f;
            if (hp >= 0 && hp < HH && x >= 1 && x <= 64)
                v = inb[(size_t)(ci0 + ci) * HWP + hp * WW + (x - 1)];
            in_t[(r * 72 + x) * CI_STR + ci] = v;
        }
        // Stage weights as [tap][co][ci]; global reads are contiguous per co.
        for (int idx = tid; idx < 9 * 16 * 64; idx += 256) {
            int t   = idx % 9;
            int ci  = (idx / 9) & 15;
            int co  = idx / 144;
            w_t[(t * 64 + co) * CI_STR + ci] =
                w[((size_t)(co0 + co) * CCH + (ci0 + ci)) * 9 + t];
        }
        __syncthreads();

        for (int t = 0; t < 9; ++t) {
            const int dh = t / 3, dw = t - 3 * (t / 3);
            const float* wt = &w_t[t * 64 * CI_STR];
            const float* it = &in_t[dh * 72 * CI_STR];
            #pragma unroll
            for (int kk = 0; kk < 4; ++kk) {
                const int ci_off = kk * 4 + 2 * half;   // K = r + 2*half
                v2f a  = ld2(&wt[(c_s * 16 + l16) * CI_STR + ci_off]);
                v2f b0 = ld2(&it[(p_s * 16 + l16 + dw) * CI_STR + ci_off]);
                acc0 = wmma_f32x4(a, b0, acc0);
                v2f b1 = ld2(&it[((p_s + 1) * 16 + l16 + dw) * CI_STR + ci_off]);
                acc1 = wmma_f32x4(a, b1, acc1);
            }
        }
    }

    // ReLU + store (D layout: row = r + 8*half, col = l16).
    float* outb = out + (size_t)b * CCH * HWP;
    #pragma unroll
    for (int r = 0; r < 8; ++r) {
        const int co = co0 + c_s * 16 + r + 8 * half;
        outb[(size_t)co * HWP + h * WW + p_s * 16 + l16]       = fmaxf(acc0[r], 0.f);
        outb[(size_t)co * HWP + h * WW + (p_s + 1) * 16 + l16] = fmaxf(acc1[r], 0.f);
    }
}

// ---------------------------------------------------------------------------
// Fused flash attention over pixel columns:
//   S = X^T X (per batch), softmax over keys n (per column m),
//   O = V * softmax(S), out = O * gamma + ref.
// Block: 128 thr (4 waves), 64 queries; each wave owns 16 queries x 256 C
// of f32 accumulators (16 v8f tiles). Keys streamed in chunks of 32.
// ---------------------------------------------------------------------------
#define Q_STR 258
#define K_STR 258
#define V_STR 34
#define P_STR 34   // P staged transposed: [m(16)][n(32)] stride 34
#define LDSF  (64 * Q_STR + 32 * K_STR + 256 * V_STR + 4 * 16 * P_STR)

__global__ __launch_bounds__(128) void attn_kernel(
    const float* __restrict__ X, const float* __restrict__ V,
    const float* __restrict__ ref, const float* __restrict__ gamma_p,
    float* __restrict__ out)
{
    extern __shared__ float lds[];
    float* q_lds = lds;                       // [64][258]  queries, [m][c]
    float* k_lds = q_lds + 64 * Q_STR;        // [32][258]  key chunk, [n][c]
    float* v_lds = k_lds + 32 * K_STR;        // [256][34]  value chunk, [c][n]
    float* p_lds = v_lds + 256 * V_STR;       // per-wave [16][34] (m-major)

    const int tid  = threadIdx.x;
    const int lane = tid & 31;
    const int wid  = tid >> 5;
    const int half = lane >> 4;
    const int l16  = lane & 15;
    const int b    = blockIdx.y;
    const int m0b  = blockIdx.x * 64;

    const float* Xb = X + (size_t)b * CCH * HWP;
    const float* Vb = V + (size_t)b * CCH * HWP;

    // Load + transpose Q tile: q_lds[m][c] = X[c][m0b+m].
    for (int idx = tid; idx < 64 * 256; idx += 128) {
        int m = idx >> 8, c = idx & 255;
        q_lds[m * Q_STR + c] = Xb[(size_t)c * HWP + m0b + m];
    }
    __syncthreads();

    v8f O[16];
    #pragma unroll
    for (int i = 0; i < 16; ++i) O[i] = (v8f){};
    float mx = -3.0e38f, lsum = 0.f;
    const int mw = wid * 16;                  // wave's query offset in block
    float* pw = p_lds + wid * 16 * P_STR;

    for (int n0 = 0; n0 < HWP; n0 += 32) {
        __syncthreads();
        for (int idx = tid; idx < 32 * 256; idx += 128) {   // K chunk (transposed)
            int n = idx >> 8, c = idx & 255;
            k_lds[n * K_STR + c] = Xb[(size_t)c * HWP + n0 + n];
        }
        for (int idx = tid; idx < 256 * 32; idx += 128) {   // V chunk (natural)
            int c = idx >> 5, n = idx & 31;
            v_lds[c * V_STR + n] = Vb[(size_t)c * HWP + n0 + n];
        }
        __syncthreads();

        // S tiles: s0 = keys[n0..n0+16), s1 = keys[n0+16..n0+32), cols = queries.
        v8f s0 = {}; v8f s1 = {};
        #pragma unroll 4
        for (int cc = 0; cc < 64; ++cc) {
            const int cofs = cc * 4 + 2 * half;
            v2f bq = ld2(&q_lds[(mw + l16) * Q_STR + cofs]);
            v2f a0 = ld2(&k_lds[l16 * K_STR + cofs]);
            s0 = wmma_f32x4(a0, bq, s0);
            v2f a1 = ld2(&k_lds[(16 + l16) * K_STR + cofs]);
            s1 = wmma_f32x4(a1, bq, s1);
        }

        // Online softmax for this lane's column (mw + l16).
        float tmax = s0[0];
        #pragma unroll
        for (int r = 0; r < 8; ++r) {
            tmax = fmaxf(tmax, s0[r]);
            tmax = fmaxf(tmax, s1[r]);
        }
        tmax = fmaxf(tmax, __shfl_xor(tmax, 16, 32));
        const float nmx   = fmaxf(mx, tmax);
        const float scale = __expf(mx - nmx);
        float p0[8], p1[8], psum = 0.f;
        #pragma unroll
        for (int r = 0; r < 8; ++r) {
            p0[r] = __expf(s0[r] - nmx);
            p1[r] = __expf(s1[r] - nmx);
            psum += p0[r] + p1[r];
        }
        psum += __shfl_xor(psum, 16, 32);
        lsum = lsum * scale + psum;
        mx = nmx;

        #pragma unroll
        for (int i = 0; i < 16; ++i)
            #pragma unroll
            for (int r = 0; r < 8; ++r) O[i][r] *= scale;

        // P -> LDS transposed [m][n]: lane's rows r,r+1 are adjacent in n,
        // so write b64 pairs. (Same-wave LDS ops are in-order.)
        #pragma unroll
        for (int r = 0; r < 8; r += 2) {
            v2f t0; t0.x = p0[r]; t0.y = p0[r + 1];
            st2(&pw[l16 * P_STR + 8 * half + r], t0);
            v2f t1; t1.x = p1[r]; t1.y = p1[r + 1];
            st2(&pw[l16 * P_STR + 16 + 8 * half + r], t1);
        }

        // O(16c x 16m) += V(16c x 32n) * P(32n x 16m)
        #pragma unroll
        for (int nn = 0; nn < 8; ++nn) {
            const int nofs = nn * 4 + 2 * half;
            v2f bp = ld2(&pw[l16 * P_STR + nofs]);
            #pragma unroll
            for (int ct = 0; ct < 16; ++ct) {
                v2f av = ld2(&v_lds[(ct * 16 + l16) * V_STR + nofs]);
                O[ct] = wmma_f32x4(av, bp, O[ct]);
            }
        }
    }

    // Epilogue: out = (O / l) * gamma + ref.
    const float inv = gamma_p[0] / lsum;
    const float* refb = ref + (size_t)b * CCH * HWP;
    float* outb       = out + (size_t)b * CCH * HWP;
    const int m = m0b + mw + l16;
    #pragma unroll
    for (int ct = 0; ct < 16; ++ct)
        #pragma unroll
        for (int r = 0; r < 8; ++r) {
            const int c = ct * 16 + r + 8 * half;
            const size_t o = (size_t)c * HWP + m;
            outb[o] = O[ct][r] * inv + refb[o];
        }
}

// ---------------------------------------------------------------------------
extern "C" void kernel_launch(void* const* d_in, const int* in_sizes, int n_in,
                              void* d_out, int out_size, void* d_ws, size_t ws_size,
                              hipStream_t stream) {
    (void)in_sizes; (void)n_in; (void)out_size; (void)ws_size;
    // d_in[0] (inputs) feeds only the dead conv -> skipped (no effect on output).
    const float* ref   = (const float*)d_in[1];
    const float* w1    = (const float*)d_in[2];
    const float* w2    = (const float*)d_in[3];
    const float* gamma = (const float*)d_in[4];
    float* out = (float*)d_out;

    float* X  = (float*)d_ws;                          // 4*256*4096 f32 = 16.8MB
    float* Vv = X + (size_t)4 * CCH * HWP;             // 16.8MB

    dim3 cgrid(CCH / 64, HH, 4);
    conv3x3_relu_kernel<<<cgrid, 256, 0, stream>>>(ref, w1, X);   // X = relu(conv1(ref))
    conv3x3_relu_kernel<<<cgrid, 256, 0, stream>>>(ref, w2, Vv);  // V = relu(conv2(ref))

    const size_t lds_bytes = (size_t)LDSF * sizeof(float);        // ~141 KB
    hipFuncSetAttribute((const void*)attn_kernel,
                        hipFuncAttributeMaxDynamicSharedMemorySize,
                        (int)lds_bytes);
    attn_kernel<<<dim3(HWP / 64, 4), 128, lds_bytes, stream>>>(X, Vv, ref, gamma, out);
}